// LossAllAtomDistances_40819369181409
// MI455X (gfx1250) — compile-verified
//
#include <hip/hip_runtime.h>
#include <stdint.h>

// ---------------------------------------------------------------------------
// LossAllAtomDistances for MI455X (gfx1250, wave32).
//  Kernel 1: KNN top-30 per residue (model + target) -> 60 edges, in d_ws.
//  Kernel 2: per-(b,n) wave computes 28x28 distance matrices per edge fully
//            inside V_WMMA_F32_16X16X4_F32 accumulators:
//              wmma1: A=(-2x,-2y,-2z,1), B=(x,y,z,n_c+eps) -> -2p.p + n_c + eps
//              wmma2 (chained C): A2=(n_r,0,0,0), B2=(1,0,0,0) -> += n_r
//            so the accumulator holds D^2+eps directly; no LDS, no shuffles.
//            All sqrt args >= eps=0.01 -> raw v_sqrt_f32, no clamps/expansions.
// ---------------------------------------------------------------------------

#define NB      2
#define NRES    768
#define NATOMS  14
#define KNN_K   30
#define NEDGE   60
#define EPS_F   0.01f

typedef __attribute__((ext_vector_type(2))) float v2f;
typedef __attribute__((ext_vector_type(8))) float v8f;

// AA20_NUM_ATOMS = {5,6,8,9,11,4,10,8,9,8,8,8,7,9,11,6,7,7,14,12}, nibble-packed
__device__ __forceinline__ int aa_count(int s) {
    const unsigned long long lo = 0x6B9788898A4B9865ULL; // s = 0..15
    const unsigned long long hi = 0x000000000000CE77ULL; // s = 16..19
    unsigned long long w = (s < 16) ? (lo >> (4 * s)) : (hi >> (4 * (s - 16)));
    return (int)(w & 15ULL);
}

// mask_ij[r]: r<14 -> r<cnt_i ; 14<=r<28 -> r-14<cnt_j ; else 0 (pad rows)
__device__ __forceinline__ float row_mask(int r, int ci, int cj) {
    if (r < NATOMS)     return (r < ci) ? 1.0f : 0.0f;
    if (r < 2 * NATOMS) return ((r - NATOMS) < cj) ? 1.0f : 0.0f;
    return 0.0f;
}

// ---------------------------------------------------------------------------
// Kernel 1: KNN. grid = (B, NRES/8), block = 256 (8 waves, 1 wave per row).
// CA coords (atom 1) of the whole batch staged in LDS (18KB + 3KB validity).
// Per-lane 24 keys (d2_bits<<10 | j); 30 rounds of wave-min extraction.
// ---------------------------------------------------------------------------
__global__ __launch_bounds__(256)
void knn_topk_kernel(const float* __restrict__ X, const float* __restrict__ Xt,
                     const int* __restrict__ C, int* __restrict__ eidx)
{
    __shared__ float ca[2][NRES][3];
    __shared__ int   cval[NRES];

    const int b   = blockIdx.x;
    const int tid = threadIdx.x;

    for (int r = tid; r < NRES; r += 256) {
        const float* p = X  + (size_t)(((b * NRES + r) * NATOMS + 1) * 3);
        const float* q = Xt + (size_t)(((b * NRES + r) * NATOMS + 1) * 3);
        ca[0][r][0] = p[0]; ca[0][r][1] = p[1]; ca[0][r][2] = p[2];
        ca[1][r][0] = q[0]; ca[1][r][1] = q[1]; ca[1][r][2] = q[2];
        cval[r] = (C[b * NRES + r] > 0) ? 1 : 0;
    }
    __syncthreads();

    const int lane = tid & 31;
    const int i    = blockIdx.y * 8 + (tid >> 5);
    const float BIGF = 1.0e9f;

    for (int s = 0; s < 2; ++s) {
        const float cx = ca[s][i][0], cy = ca[s][i][1], cz = ca[s][i][2];
        const int   vi = cval[i];

        unsigned long long keys[NRES / 32];
#pragma unroll
        for (int t = 0; t < NRES / 32; ++t) {
            const int j = t * 32 + lane;
            const float dx = ca[s][j][0] - cx;
            const float dy = ca[s][j][1] - cy;
            const float dz = ca[s][j][2] - cz;
            float d2 = dx * dx + dy * dy + dz * dz;
            if (!(vi && cval[j])) d2 = BIGF;
            // d2 >= 0 -> float bits are order-preserving as unsigned
            keys[t] = ((unsigned long long)__float_as_uint(d2) << 10) |
                      (unsigned long long)(unsigned)j;
        }

#pragma unroll 1
        for (int k = 0; k < KNN_K; ++k) {
            unsigned long long m = ~0ULL;
#pragma unroll
            for (int t = 0; t < NRES / 32; ++t) m = (keys[t] < m) ? keys[t] : m;
#pragma unroll
            for (int off = 16; off > 0; off >>= 1) {
                unsigned long long o = __shfl_xor(m, off, 32);
                m = (o < m) ? o : m;
            }
            if (lane == 0)
                eidx[(b * NRES + i) * NEDGE + s * KNN_K + k] = (int)(m & 1023ULL);
#pragma unroll
            for (int t = 0; t < NRES / 32; ++t)
                if (keys[t] == m) keys[t] = ~0ULL;
        }
    }
}

// ---------------------------------------------------------------------------
// Kernel 2: loss. One wave per (b, n).
// P rows: 0..13 = atoms of residue i, 14..27 = atoms of residue j, 28..31 pad.
// Lane l owns rows lr=l&15 (tile0) and 16+lr (tile1) -> both norms local.
// WMMA f32 16x16x4 operand image: lanes 0-15 hold (K0,K1), 16-31 hold (K2,K3).
// Symmetric D: compute tiles (0,0), (0,1)x2, (1,1) only.
// ---------------------------------------------------------------------------
__global__ __launch_bounds__(256)
void allatom_loss_kernel(const float* __restrict__ X, const float* __restrict__ Xt,
                         const int* __restrict__ C, const int* __restrict__ S,
                         const int* __restrict__ eidx, float* __restrict__ out)
{
    const int gw   = (int)((blockIdx.x * blockDim.x + threadIdx.x) >> 5);
    const int lane = threadIdx.x & 31;
    if (gw >= NB * NRES) return;              // wave-uniform (grid is exact)
    const int b = gw / NRES;
    const int i = gw % NRES;

    const int  lr     = lane & 15;
    const bool hiHalf = (lane >= 16);

    const int cnt_i = (C[b * NRES + i] > 0) ? aa_count(S[b * NRES + i]) : 0;

    // Edge-invariant: atoms of residue i for rows lr<14 (clamped, uncond load).
    const int aI    = (lr < NATOMS) ? lr : 0;
    const int baseI = ((b * NRES + i) * NATOMS + aI) * 3;
    const float ixm = X[baseI + 0],  iym = X[baseI + 1],  izm = X[baseI + 2];
    const float ixt = Xt[baseI + 0], iyt = Xt[baseI + 1], izt = Xt[baseI + 2];

    // Constant B2 operand: column vector of ones in K-slot 0 -> adds A2's n_r.
    v2f B2; B2[0] = hiHalf ? 0.0f : 1.0f; B2[1] = 0.0f;

    float accNum = 0.f;
    float accDen = 0.f;
    const int* ep = eidx + (b * NRES + i) * NEDGE;
    const v8f  cz = {};

#pragma unroll 1
    for (int e = 0; e < NEDGE; ++e) {
        const int j     = ep[e];
        const int cnt_j = (C[b * NRES + j] > 0) ? aa_count(S[b * NRES + j]) : 0;
        const int baseJ = (b * NRES + j) * NATOMS * 3;

        // Row rLo = lr: i atom lr (lr<14) or j atom lr-14. Clamped uncond load.
        const int aLo = (lr >= NATOMS) ? (lr - NATOMS) : 0;
        const int oLo = baseJ + aLo * 3;
        const float jlxm = X[oLo + 0],  jlym = X[oLo + 1],  jlzm = X[oLo + 2];
        const float jlxt = Xt[oLo + 0], jlyt = Xt[oLo + 1], jlzt = Xt[oLo + 2];
        const bool  loI  = (lr < NATOMS);
        const float lxm = loI ? ixm : jlxm, lym = loI ? iym : jlym, lzm = loI ? izm : jlzm;
        const float lxt = loI ? ixt : jlxt, lyt = loI ? iyt : jlyt, lzt = loI ? izt : jlzt;

        // Row rHi = 16+lr: j atom lr+2 (lr<12), zero-pad otherwise.
        const int aHi = (lr + 2 < NATOMS) ? (lr + 2) : (NATOMS - 1);
        const int oHi = baseJ + aHi * 3;
        const bool  hv  = (lr < NATOMS - 2);
        const float hxm = hv ? X[oHi + 0] : 0.f,  hym = hv ? X[oHi + 1] : 0.f,  hzm = hv ? X[oHi + 2] : 0.f;
        const float hxt = hv ? Xt[oHi + 0] : 0.f, hyt = hv ? Xt[oHi + 1] : 0.f, hzt = hv ? Xt[oHi + 2] : 0.f;

        // Local norms of both owned rows (no shuffles needed).
        const float nLo_m = lxm * lxm + lym * lym + lzm * lzm;
        const float nHi_m = hxm * hxm + hym * hym + hzm * hzm;
        const float nLo_t = lxt * lxt + lyt * lyt + lzt * lzt;
        const float nHi_t = hxt * hxt + hyt * hyt + hzt * hzt;

        // Operands. A: (-2x,-2y | -2z, 1). B: (x, y | z, n+eps). A2: (n,0 | 0,0).
        v2f A0m, A1m, B0m, B1m, N0m, N1m;
        v2f A0t, A1t, B0t, B1t, N0t, N1t;
        A0m[0] = hiHalf ? -2.f * lzm : -2.f * lxm;  A0m[1] = hiHalf ? 1.f : -2.f * lym;
        A1m[0] = hiHalf ? -2.f * hzm : -2.f * hxm;  A1m[1] = hiHalf ? 1.f : -2.f * hym;
        B0m[0] = hiHalf ? lzm : lxm;                B0m[1] = hiHalf ? nLo_m + EPS_F : lym;
        B1m[0] = hiHalf ? hzm : hxm;                B1m[1] = hiHalf ? nHi_m + EPS_F : hym;
        N0m[0] = hiHalf ? 0.f : nLo_m;              N0m[1] = 0.f;
        N1m[0] = hiHalf ? 0.f : nHi_m;              N1m[1] = 0.f;
        A0t[0] = hiHalf ? -2.f * lzt : -2.f * lxt;  A0t[1] = hiHalf ? 1.f : -2.f * lyt;
        A1t[0] = hiHalf ? -2.f * hzt : -2.f * hxt;  A1t[1] = hiHalf ? 1.f : -2.f * hyt;
        B0t[0] = hiHalf ? lzt : lxt;                B0t[1] = hiHalf ? nLo_t + EPS_F : lyt;
        B1t[0] = hiHalf ? hzt : hxt;                B1t[1] = hiHalf ? nHi_t + EPS_F : hyt;
        N0t[0] = hiHalf ? 0.f : nLo_t;              N0t[1] = 0.f;
        N1t[0] = hiHalf ? 0.f : nHi_t;              N1t[1] = 0.f;

#pragma unroll
        for (int ti = 0; ti < 2; ++ti) {
            const v2f Am = ti ? A1m : A0m;  const v2f At = ti ? A1t : A0t;
            const v2f Nm = ti ? N1m : N0m;  const v2f Nt = ti ? N1t : N0t;

            // Row masks depend only on ti (hoisted out of the tj loop).
            float rmask[8];
#pragma unroll
            for (int v = 0; v < 8; ++v)
                rmask[v] = row_mask(ti * 16 + v + (hiHalf ? 8 : 0), cnt_i, cnt_j);

#pragma unroll
            for (int tj = ti; tj < 2; ++tj) {
                const v2f Bm = tj ? B1m : B0m;
                const v2f Bt = tj ? B1t : B0t;

                // Accumulator = -2 p_r.p_c + n_c + eps, then += n_r  => D^2+eps
                v8f gm = __builtin_amdgcn_wmma_f32_16x16x4_f32(
                    false, Am, false, Bm, (short)0, cz, false, false);
                gm = __builtin_amdgcn_wmma_f32_16x16x4_f32(
                    false, Nm, false, B2, (short)0, gm, false, false);
                v8f gt = __builtin_amdgcn_wmma_f32_16x16x4_f32(
                    false, At, false, Bt, (short)0, cz, false, false);
                gt = __builtin_amdgcn_wmma_f32_16x16x4_f32(
                    false, Nt, false, B2, (short)0, gt, false, false);

                const float w    = (ti != tj) ? 2.0f : 1.0f; // symmetric halves
                const int   c    = tj * 16 + lr;
                const float mcol = row_mask(c, cnt_i, cnt_j);

                float part = 0.f;
#pragma unroll
                for (int v = 0; v < 8; ++v) {
                    // args >= eps = 0.01 -> raw v_sqrt_f32 is safe & exact enough
                    const float dm = __builtin_amdgcn_sqrtf(gm[v]);
                    const float dt = __builtin_amdgcn_sqrtf(gt[v]);
                    const float df = dm - dt;
                    const float ls = __builtin_amdgcn_sqrtf(df * df + EPS_F);
                    part += rmask[v] * ls;
                }
                accNum += w * mcol * part;
            }
        }
        const float t = (float)(cnt_i + cnt_j);  // sum(mask_ij) per edge
        accDen += t * t;                          // den contribution (analytic)
    }

#pragma unroll
    for (int off = 16; off > 0; off >>= 1) accNum += __shfl_xor(accNum, off, 32);
    if (lane == 0) out[b * NRES + i] = accNum / (accDen + EPS_F);
}

// ---------------------------------------------------------------------------
extern "C" void kernel_launch(void* const* d_in, const int* in_sizes, int n_in,
                              void* d_out, int out_size, void* d_ws, size_t ws_size,
                              hipStream_t stream) {
    const float* X  = (const float*)d_in[0];
    const float* Xt = (const float*)d_in[1];
    const int*   C  = (const int*)d_in[2];
    const int*   S  = (const int*)d_in[3];

    int* eidx = (int*)d_ws;  // NB*NRES*NEDGE ints = 360 KB

    dim3 g1(NB, NRES / 8);
    knn_topk_kernel<<<g1, 256, 0, stream>>>(X, Xt, C, eidx);

    const int totalWaves   = NB * NRES;                // 1536
    const int wavesPerBlk  = 256 / 32;                 // 8
    const int blocks       = totalWaves / wavesPerBlk; // 192
    allatom_loss_kernel<<<blocks, 256, 0, stream>>>(X, Xt, C, S, eidx,
                                                    (float*)d_out);
}